// GNN_66434554134706
// MI455X (gfx1250) — compile-verified
//
#include <hip/hip_runtime.h>
#include <hip/hip_bf16.h>

typedef float v2f __attribute__((ext_vector_type(2)));
typedef float v8f __attribute__((ext_vector_type(8)));

#define EMB 32
#define NREL 8
#define NLAYER 5

// ---------------------------------------------------------------------------
// One wave computes a 16x32 output tile:  out[row0:row0+16, 0:32] =
//   in[row0:row0+16, 0:32] @ W[32,32] (+bias) using V_WMMA_F32_16X16X4_F32.
// K=32 -> 8 chained WMMAs per 16-col half. All guards are wave-uniform so
// EXEC is all-ones at the WMMA (ISA requirement).
// ---------------------------------------------------------------------------
__device__ __forceinline__ void gemm_tile16(const float* __restrict__ in, int row0, int rows,
                                            const float* __restrict__ W,
                                            const float* __restrict__ bias,
                                            float* __restrict__ out, long ostride, bool relu) {
    int lane = threadIdx.x & 31;
    if (rows == 16) {
        int half = lane >> 4;     // 0: lanes 0-15, 1: lanes 16-31
        int l15  = lane & 15;
        v8f c0, c1;
        float b0 = bias ? bias[l15]      : 0.0f;
        float b1 = bias ? bias[16 + l15] : 0.0f;
#pragma unroll
        for (int v = 0; v < 8; ++v) { c0[v] = b0; c1[v] = b1; }
        const float* arow = in + (size_t)row0 * EMB;
#pragma unroll
        for (int k = 0; k < 32; k += 4) {
            // A 16x4 tile: VGPR0 = K=k+2*half, VGPR1 = K=k+2*half+1 (per lane row l15)
            v2f a, bA, bB;
            const float* ap = arow + (size_t)l15 * EMB + k + half * 2;
            a[0] = ap[0];
            a[1] = ap[1];
            // B 4x16 tiles (cols 0-15 and 16-31): row-striped across lanes
            const float* bp = W + (k + half * 2) * EMB + l15;
            bA[0] = bp[0];  bA[1] = bp[EMB];        // K=k+2h, k+2h+1 ; col l15
            bB[0] = bp[16]; bB[1] = bp[EMB + 16];   // same, cols 16-31
            c0 = __builtin_amdgcn_wmma_f32_16x16x4_f32(false, a, false, bA, (short)0, c0, false, false);
            c1 = __builtin_amdgcn_wmma_f32_16x16x4_f32(false, a, false, bB, (short)0, c1, false, false);
        }
        int rbase = half * 8;   // C layout: VGPR v -> row v (lanes 0-15) / v+8 (lanes 16-31)
#pragma unroll
        for (int v = 0; v < 8; ++v) {
            float x0 = c0[v], x1 = c1[v];
            if (relu) { x0 = fmaxf(x0, 0.0f); x1 = fmaxf(x1, 0.0f); }
            out[(size_t)(row0 + rbase + v) * ostride + l15]      = x0;
            out[(size_t)(row0 + rbase + v) * ostride + 16 + l15] = x1;
        }
    } else {
        // partial-tile scalar fallback (not hit for N=50000, kept for safety)
        for (int r = 0; r < rows; ++r) {
            float acc = bias ? bias[lane] : 0.0f;
            const float* hr = in + (size_t)(row0 + r) * EMB;
            for (int i = 0; i < EMB; ++i) acc += hr[i] * W[i * EMB + lane];
            if (relu) acc = fmaxf(acc, 0.0f);
            out[(size_t)(row0 + r) * ostride + lane] = acc;
        }
    }
}

// h0 = relu(x @ fc_W + fc_b)   (also generic single-GEMM-over-nodes kernel)
__global__ void node_gemm_kernel(const float* __restrict__ in, const float* __restrict__ W,
                                 const float* __restrict__ bias, float* __restrict__ out,
                                 int n, int relu) {
    int tile = blockIdx.x * (blockDim.x >> 5) + (threadIdx.x >> 5);
    int row0 = tile * 16;
    if (row0 >= n) return;
    int rows = n - row0; if (rows > 16) rows = 16;
    gemm_tile16(in, row0, rows, W, bias, out, EMB, relu != 0);
}

// Per-layer batch of 43 GEMMs, selected by blockIdx.y:
//  g in [0,8)  : P[g]      = h @ rgcn_W[l,g]
//  g == 8      : rgcnout   = h @ rgcn_root[l] + rgcn_bias[l]
//  g == 9      : nnrootout = h @ nn_root[l]
//  g == 10     : u         = h @ Bmat          (Bmat[i,o] = mlp_b2[i*32+o])
//  g in [11,43): M[:,k,:]  = h @ W2_k, k=g-11  (W2_k[i,o] = mlp_W2[k*1024+i*32+o])
__global__ void layer_gemms_kernel(const float* __restrict__ h,
                                   const float* __restrict__ rgcnW_l,
                                   const float* __restrict__ rgcn_root_l,
                                   const float* __restrict__ rgcn_bias_l,
                                   const float* __restrict__ nn_root_l,
                                   const float* __restrict__ mlp_W2,
                                   const float* __restrict__ mlp_b2,
                                   float* __restrict__ P, float* __restrict__ rgcnout,
                                   float* __restrict__ nnrootout, float* __restrict__ u,
                                   float* __restrict__ M, int n) {
    int tile = blockIdx.x * (blockDim.x >> 5) + (threadIdx.x >> 5);
    int row0 = tile * 16;
    if (row0 >= n) return;
    int rows = n - row0; if (rows > 16) rows = 16;
    int g = blockIdx.y;

    const float* W;
    const float* bias = nullptr;
    float* out;
    long stride = EMB;
    if (g < 8) {
        W = rgcnW_l + (size_t)g * EMB * EMB;
        out = P + (size_t)g * n * EMB;
    } else if (g == 8) {
        W = rgcn_root_l; bias = rgcn_bias_l; out = rgcnout;
    } else if (g == 9) {
        W = nn_root_l; out = nnrootout;
    } else if (g == 10) {
        W = mlp_b2; out = u;
    } else {
        int k = g - 11;
        W = mlp_W2 + (size_t)k * EMB * EMB;
        out = M + (size_t)k * EMB;     // element (j,k,o) at M[j*1024 + k*32 + o]
        stride = EMB * EMB;
    }
    gemm_tile16(h, row0, rows, W, bias, out, stride, false);
}

// z_e = relu(dist_e * W1[0,:] + W1[1+type_e,:] + b1)
__global__ void edge_z_kernel(const float* __restrict__ dist, const int* __restrict__ etype,
                              const float* __restrict__ W1, const float* __restrict__ b1,
                              float* __restrict__ z, int nE) {
    size_t idx = (size_t)blockIdx.x * blockDim.x + threadIdx.x;
    if (idx >= (size_t)nE * EMB) return;
    int e = (int)(idx >> 5), j = (int)(idx & 31);
    float v = dist[e] * W1[j] + W1[(size_t)(1 + etype[e]) * EMB + j] + b1[j];
    z[idx] = fmaxf(v, 0.0f);
}

__global__ void zero_f32_kernel(float* __restrict__ p, size_t cnt) {
    size_t i = (size_t)blockIdx.x * blockDim.x + threadIdx.x;
    if (i < cnt) p[i] = 0.0f;
}
__global__ void zero_i32_kernel(int* __restrict__ p, size_t cnt) {
    size_t i = (size_t)blockIdx.x * blockDim.x + threadIdx.x;
    if (i < cnt) p[i] = 0;
}

__global__ void count_deg_kernel(const int* __restrict__ dst, const int* __restrict__ etype,
                                 int* __restrict__ degi, int* __restrict__ relcnti, int nE) {
    int e = blockIdx.x * blockDim.x + threadIdx.x;
    if (e >= nE) return;
    int d = dst[e], t = etype[e];
    atomicAdd(&degi[d], 1);
    atomicAdd(&relcnti[(size_t)d * NREL + t], 1);
}

__global__ void finalize_inv_kernel(const int* __restrict__ degi, const int* __restrict__ relcnti,
                                    float* __restrict__ inv_deg, float* __restrict__ inv_relcnt,
                                    int n) {
    int i = blockIdx.x * blockDim.x + threadIdx.x;
    if (i >= n) return;
    inv_deg[i] = 1.0f / fmaxf((float)degi[i], 1.0f);
#pragma unroll
    for (int r = 0; r < NREL; ++r)
        inv_relcnt[(size_t)i * NREL + r] = 1.0f / fmaxf((float)relcnti[(size_t)i * NREL + r], 1.0f);
}

// One wave per edge, 32 lanes = 32 features.
//   rgcn: rgcnout[dst] += P[type][src] * inv_relcnt[dst,type]
//   nn  : agg[dst]     += z_e @ M[src] + u[src]
__global__ void edge_scatter_kernel(const int* __restrict__ src, const int* __restrict__ dst,
                                    const int* __restrict__ etype, const float* __restrict__ z,
                                    const float* __restrict__ P, const float* __restrict__ M,
                                    const float* __restrict__ u, const float* __restrict__ inv_relcnt,
                                    float* __restrict__ rgcnout, float* __restrict__ agg,
                                    int nE, int n) {
    int e = blockIdx.x * (blockDim.x >> 5) + (threadIdx.x >> 5);
    if (e >= nE) return;                       // wave-uniform
    int lane = threadIdx.x & 31;
    int s = src[e], d = dst[e], t = etype[e];

    float pv = P[((size_t)t * n + s) * EMB + lane] * inv_relcnt[(size_t)d * NREL + t];
    __hip_atomic_fetch_add(&rgcnout[(size_t)d * EMB + lane], pv,
                           __ATOMIC_RELAXED, __HIP_MEMORY_SCOPE_AGENT);

    float zl = z[(size_t)e * EMB + lane];
    float acc = u[(size_t)s * EMB + lane];
    const float* Ms = M + (size_t)s * EMB * EMB;   // contiguous 4 KB block per src node
#pragma unroll
    for (int k = 0; k < EMB; ++k) {
        float zk = __shfl(zl, k, 32);
        acc += zk * Ms[k * EMB + lane];
    }
    __hip_atomic_fetch_add(&agg[(size_t)d * EMB + lane], acc,
                           __ATOMIC_RELAXED, __HIP_MEMORY_SCOPE_AGENT);
}

// h_next = h + relu(rgcnout) + relu(nnrootout + agg*inv_deg + nn_bias)
__global__ void combine_kernel(const float* __restrict__ h, const float* __restrict__ rgcnout,
                               const float* __restrict__ nnrootout, const float* __restrict__ agg,
                               const float* __restrict__ inv_deg, const float* __restrict__ nn_bias_l,
                               float* __restrict__ hout, int n) {
    size_t i = (size_t)blockIdx.x * blockDim.x + threadIdx.x;
    if (i >= (size_t)n * EMB) return;
    int node = (int)(i >> 5), j = (int)(i & 31);
    float hd = fmaxf(rgcnout[i], 0.0f);
    float hc = fmaxf(nnrootout[i] + agg[i] * inv_deg[node] + nn_bias_l[j], 0.0f);
    hout[i] = h[i] + hd + hc;
}

extern "C" void kernel_launch(void* const* d_in, const int* in_sizes, int n_in,
                              void* d_out, int out_size, void* d_ws, size_t ws_size,
                              hipStream_t stream) {
    const float* x         = (const float*)d_in[0];
    const int*   ei        = (const int*)  d_in[1];
    const int*   etype     = (const int*)  d_in[2];
    const float* edist     = (const float*)d_in[3];
    const float* fc_W      = (const float*)d_in[4];
    const float* fc_b      = (const float*)d_in[5];
    const float* rgcn_W    = (const float*)d_in[6];
    const float* rgcn_root = (const float*)d_in[7];
    const float* rgcn_bias = (const float*)d_in[8];
    const float* mlp_W1    = (const float*)d_in[9];
    const float* mlp_b1    = (const float*)d_in[10];
    const float* mlp_W2    = (const float*)d_in[11];
    const float* mlp_b2    = (const float*)d_in[12];
    const float* nn_root   = (const float*)d_in[13];
    const float* nn_bias   = (const float*)d_in[14];

    const int N = in_sizes[0] / EMB;   // 50000
    const int E = in_sizes[2];         // 400000
    const int* srcp = ei;
    const int* dstp = ei + E;

    // ---- workspace layout (floats) ----
    float* w = (float*)d_ws;
    size_t o = 0;
    float* z          = w + o; o += (size_t)E * EMB;            // per-edge MLP hidden
    float* P          = w + o; o += (size_t)NREL * N * EMB;     // per-relation projections
    float* M          = w + o; o += (size_t)N * EMB * EMB;      // per-node h @ W2_k
    float* u          = w + o; o += (size_t)N * EMB;            // per-node h @ Bmat
    float* rgcnout    = w + o; o += (size_t)N * EMB;
    float* nnrootout  = w + o; o += (size_t)N * EMB;
    float* agg        = w + o; o += (size_t)N * EMB;
    float* hA         = w + o; o += (size_t)N * EMB;
    float* hB         = w + o; o += (size_t)N * EMB;
    float* inv_deg    = w + o; o += (size_t)N;
    float* inv_relcnt = w + o; o += (size_t)N * NREL;
    int*   degi       = (int*)(w + o); o += (size_t)N;
    int*   relcnti    = (int*)(w + o); o += (size_t)N * NREL;

    const int T = 256;
    int ntiles = (N + 15) / 16;
    dim3 gemmBlk(128);                                   // 4 waves / block
    int  gemmGx = (ntiles + 3) / 4;

    // ---- one-time precompute ----
    zero_i32_kernel<<<(unsigned)(((size_t)N * (NREL + 1) + T - 1) / T), T, 0, stream>>>(degi, (size_t)N * (NREL + 1));
    count_deg_kernel<<<(E + T - 1) / T, T, 0, stream>>>(dstp, etype, degi, relcnti, E);
    finalize_inv_kernel<<<(N + T - 1) / T, T, 0, stream>>>(degi, relcnti, inv_deg, inv_relcnt, N);
    edge_z_kernel<<<(unsigned)(((size_t)E * EMB + T - 1) / T), T, 0, stream>>>(edist, etype, mlp_W1, mlp_b1, z, E);

    // h0 = relu(x @ fc_W + fc_b)
    node_gemm_kernel<<<gemmGx, gemmBlk, 0, stream>>>(x, fc_W, fc_b, hA, N, 1);

    float* hcur = hA;
    for (int l = 0; l < NLAYER; ++l) {
        layer_gemms_kernel<<<dim3(gemmGx, 11 + EMB), gemmBlk, 0, stream>>>(
            hcur,
            rgcn_W + (size_t)l * NREL * EMB * EMB,
            rgcn_root + (size_t)l * EMB * EMB,
            rgcn_bias + (size_t)l * EMB,
            nn_root + (size_t)l * EMB * EMB,
            mlp_W2, mlp_b2,
            P, rgcnout, nnrootout, u, M, N);

        zero_f32_kernel<<<(unsigned)(((size_t)N * EMB + T - 1) / T), T, 0, stream>>>(agg, (size_t)N * EMB);

        edge_scatter_kernel<<<(E + 7) / 8, 256, 0, stream>>>(
            srcp, dstp, etype, z, P, M, u, inv_relcnt, rgcnout, agg, E, N);

        float* hnext = (l == NLAYER - 1) ? (float*)d_out : ((l & 1) ? hA : hB);
        combine_kernel<<<(unsigned)(((size_t)N * EMB + T - 1) / T), T, 0, stream>>>(
            hcur, rgcnout, nnrootout, agg, inv_deg, nn_bias + (size_t)l * EMB, hnext, N);
        hcur = hnext;
    }
}